// NMSDecoder_6897717477870
// MI455X (gfx1250) — compile-verified
//
#include <hip/hip_runtime.h>

// Problem constants (fixed by the reference).
#define BATCH   512
#define NVARS   16384
#define NCHECKS 8192
#define DCHK    6
#define NEDGES  (NCHECKS * DCHK)   // 49152
#define NITERS  5

// ---- CDNA5 async global->LDS DMA (ASYNCcnt path), guarded for portability --
#if defined(__has_builtin)
#if __has_builtin(__builtin_amdgcn_global_load_async_to_lds_b32) && \
    __has_builtin(__builtin_amdgcn_s_wait_asynccnt)
#define USE_ASYNC_LDS 1
#endif
#endif
#ifndef USE_ASYNC_LDS
#define USE_ASYNC_LDS 0
#endif

#if USE_ASYNC_LDS
// Parameter types per hipcc diagnostic: arg0 is AS(1) int* ("__device__ int*"),
// arg1 is the LDS-side pointer (AS(3) int*).
typedef __attribute__((address_space(1))) int gbl_int;
typedef __attribute__((address_space(3))) int lds_int;
#endif

// ---------------------------------------------------------------------------
// 1) Transpose channel LLRs [B, V] -> chT [V, B]  (LDS tile transpose).
//    Load phase is a pure global->LDS copy: use async-to-LDS DMA when the
//    toolchain exposes it (per-lane global addr -> per-lane LDS addr,
//    completion via s_wait_asynccnt 0 before the workgroup barrier).
// ---------------------------------------------------------------------------
__global__ void k_transpose_in(const float* __restrict__ in, float* __restrict__ out) {
    __shared__ float tile[32][33];
    const int v0 = blockIdx.x * 32;
    const int b0 = blockIdx.y * 32;
#if USE_ASYNC_LDS
    for (int i = threadIdx.y; i < 32; i += 8) {
        const float* g = in + (size_t)(b0 + i) * NVARS + v0 + threadIdx.x;
        __builtin_amdgcn_global_load_async_to_lds_b32(
            (gbl_int*)g, (lds_int*)&tile[i][threadIdx.x], 0, 0);
    }
    __builtin_amdgcn_s_wait_asynccnt(0);
#else
    for (int i = threadIdx.y; i < 32; i += 8)
        tile[i][threadIdx.x] = in[(size_t)(b0 + i) * NVARS + v0 + threadIdx.x];
#endif
    __syncthreads();
    for (int i = threadIdx.y; i < 32; i += 8)
        out[(size_t)(v0 + i) * BATCH + b0 + threadIdx.x] = tile[threadIdx.x][i];
}

// ---------------------------------------------------------------------------
// 2) Build inverse CSR: var -> its 3 edge ids. Atomic slot assignment is
//    nondeterministic, so we sort the 3 entries -> deterministic result.
// ---------------------------------------------------------------------------
__global__ void k_count(const int* __restrict__ Hc, int* __restrict__ cnt,
                        int* __restrict__ v2e) {
    const int e = blockIdx.x * 256 + threadIdx.x;
    if (e < NEDGES) {
        const int v = Hc[e];
        const int s = atomicAdd(&cnt[v], 1);
        if (s < 3) v2e[v * 4 + s] = e;
    }
}

__global__ void k_sort3(int* __restrict__ v2e) {
    const int v = blockIdx.x * 256 + threadIdx.x;
    if (v < NVARS) {
        int a = v2e[v * 4 + 0], b = v2e[v * 4 + 1], c = v2e[v * 4 + 2], t;
        if (a > b) { t = a; a = b; b = t; }
        if (b > c) { t = b; b = c; c = t; }
        if (a > b) { t = a; a = b; b = t; }
        v2e[v * 4 + 0] = a; v2e[v * 4 + 1] = b;
        v2e[v * 4 + 2] = c; v2e[v * 4 + 3] = 0;
    }
}

// ---------------------------------------------------------------------------
// 3) Variable-node update, in place on msg[E, B]:
//    msg holds c2v on entry, v2c on exit.  v2c[e] = ch[v] + sum(c2v) - c2v[e].
//    First iteration: c2v == 0  ->  v2c[e] = ch[v].
//    One wave row (32 lanes) x float4 = 128 batch elements; v is y-uniform.
// ---------------------------------------------------------------------------
__global__ void k_var_update(const float* __restrict__ chT, float* __restrict__ msg,
                             const int* __restrict__ v2e, const int first) {
    const int b = (blockIdx.x * 32 + threadIdx.x) * 4;
    const int v = blockIdx.y * 8 + threadIdx.y;
    const int4 e = *reinterpret_cast<const int4*>(v2e + v * 4);
    const size_t o0 = (size_t)e.x * BATCH + b;
    const size_t o1 = (size_t)e.y * BATCH + b;
    const size_t o2 = (size_t)e.z * BATCH + b;
    const float4 ch = *reinterpret_cast<const float4*>(chT + (size_t)v * BATCH + b);
    if (first) {
        *reinterpret_cast<float4*>(msg + o0) = ch;
        *reinterpret_cast<float4*>(msg + o1) = ch;
        *reinterpret_cast<float4*>(msg + o2) = ch;
    } else {
        const float4 c0 = *reinterpret_cast<const float4*>(msg + o0);
        const float4 c1 = *reinterpret_cast<const float4*>(msg + o1);
        const float4 c2 = *reinterpret_cast<const float4*>(msg + o2);
        float4 s;
        s.x = ch.x + c0.x + c1.x + c2.x;
        s.y = ch.y + c0.y + c1.y + c2.y;
        s.z = ch.z + c0.z + c1.z + c2.z;
        s.w = ch.w + c0.w + c1.w + c2.w;
        *reinterpret_cast<float4*>(msg + o0) =
            make_float4(s.x - c0.x, s.y - c0.y, s.z - c0.z, s.w - c0.w);
        *reinterpret_cast<float4*>(msg + o1) =
            make_float4(s.x - c1.x, s.y - c1.y, s.z - c1.z, s.w - c1.w);
        *reinterpret_cast<float4*>(msg + o2) =
            make_float4(s.x - c2.x, s.y - c2.y, s.z - c2.z, s.w - c2.w);
    }
}

// ---------------------------------------------------------------------------
// 4) Check-node min-sum update, in place on msg (v2c -> c2v), scaled by w.
//    Edges of check c are contiguous: rows [6c, 6c+6) of msg[E, B].
//    Matches reference semantics: zero input annihilates the whole check;
//    unique-min edge gets min2, everything else min1; extrinsic sign = XOR
//    of all sign bits except own.
// ---------------------------------------------------------------------------
__device__ __forceinline__ void minsum6(const float m[DCHK], const float w, float o[DCHK]) {
    unsigned xs = 0u;
    float a[DCHK];
    float min1 = __builtin_inff(), min2 = __builtin_inff();
    int   hz = 0;
#pragma unroll
    for (int j = 0; j < DCHK; ++j) {
        const unsigned bits = __float_as_uint(m[j]);
        xs ^= (bits & 0x80000000u);
        const float aj = __uint_as_float(bits & 0x7fffffffu);
        a[j] = aj;
        hz |= (aj == 0.0f);
        if (aj < min1)      { min2 = min1; min1 = aj; }
        else if (aj < min2) { min2 = aj; }
    }
#pragma unroll
    for (int j = 0; j < DCHK; ++j) {
        const float omin = (a[j] == min1) ? min2 : min1;
        const unsigned sj = __float_as_uint(m[j]) & 0x80000000u;
        const float val = __uint_as_float((xs ^ sj) | __float_as_uint(omin));
        o[j] = hz ? 0.0f : val * w;
    }
}

__global__ void k_check_update(float* __restrict__ msg, const float* __restrict__ wts,
                               const int iter) {
    const int c = blockIdx.y;                       // one check per blockIdx.y
    const int b = (blockIdx.x * 128 + threadIdx.x) * 4;
    const float w = wts[iter];
    float* p = msg + (size_t)c * DCHK * BATCH + b;

    if (c + 1 < NCHECKS)                            // gfx1250 global_prefetch_b8
        __builtin_prefetch(p + (size_t)DCHK * BATCH, 0, 1);

    float mm[4][DCHK];
#pragma unroll
    for (int j = 0; j < DCHK; ++j) {
        const float4 t = *reinterpret_cast<const float4*>(p + (size_t)j * BATCH);
        mm[0][j] = t.x; mm[1][j] = t.y; mm[2][j] = t.z; mm[3][j] = t.w;
    }
    float oo[4][DCHK];
#pragma unroll
    for (int k = 0; k < 4; ++k) minsum6(mm[k], w, oo[k]);
#pragma unroll
    for (int j = 0; j < DCHK; ++j) {
        *reinterpret_cast<float4*>(p + (size_t)j * BATCH) =
            make_float4(oo[0][j], oo[1][j], oo[2][j], oo[3][j]);
    }
}

// ---------------------------------------------------------------------------
// 5) Final: out[b, v] = ch[v, b] + sum of v's 3 c2v messages, fused with the
//    transpose back to [B, V] through an LDS tile.
// ---------------------------------------------------------------------------
__global__ void k_final(const float* __restrict__ chT, const float* __restrict__ msg,
                        const int* __restrict__ v2e, float* __restrict__ out) {
    __shared__ float tile[32][33];
    const int v0 = blockIdx.y * 32;
    const int b0 = blockIdx.x * 32;
    for (int i = threadIdx.y; i < 32; i += 8) {
        const int v = v0 + i;
        const int b = b0 + threadIdx.x;
        const int4 e = *reinterpret_cast<const int4*>(v2e + v * 4);
        const float s = chT[(size_t)v * BATCH + b]
                      + msg[(size_t)e.x * BATCH + b]
                      + msg[(size_t)e.y * BATCH + b]
                      + msg[(size_t)e.z * BATCH + b];
        tile[i][threadIdx.x] = s;
    }
    __syncthreads();
    for (int i = threadIdx.y; i < 32; i += 8)
        out[(size_t)(b0 + i) * NVARS + v0 + threadIdx.x] = tile[threadIdx.x][i];
}

// ---------------------------------------------------------------------------
// Host orchestration. Workspace layout:
//   chT  : NVARS*BATCH  f32   (32 MB)
//   msg  : NEDGES*BATCH f32   (96 MB)   -- single in-place message array
//   v2e  : NVARS*4      i32   (256 KB)
//   cnt  : NVARS        i32   (64 KB)
// Total ~128.3 MB -> fits the 192 MB global L2; HBM sees ~32 MB in + 32 MB out.
// ---------------------------------------------------------------------------
extern "C" void kernel_launch(void* const* d_in, const int* in_sizes, int n_in,
                              void* d_out, int out_size, void* d_ws, size_t ws_size,
                              hipStream_t stream) {
    (void)in_sizes; (void)n_in; (void)out_size; (void)ws_size;

    const float* channel = (const float*)d_in[0];   // [B, NVARS] f32
    const float* weights = (const float*)d_in[1];   // [NITERS]   f32
    // d_in[2] = H_rows (== e / DCHK by construction; not needed)
    const int*   Hcols   = (const int*)d_in[3];     // [NEDGES]   i32
    float*       out     = (float*)d_out;

    char* ws = (char*)d_ws;
    float* chT = (float*)ws;
    float* msg = (float*)(ws + sizeof(float) * (size_t)NVARS * BATCH);
    int*   v2e = (int*)(ws + sizeof(float) * ((size_t)NVARS * BATCH + (size_t)NEDGES * BATCH));
    int*   cnt = v2e + NVARS * 4;

    (void)hipMemsetAsync(cnt, 0, NVARS * sizeof(int), stream);

    k_transpose_in<<<dim3(NVARS / 32, BATCH / 32), dim3(32, 8), 0, stream>>>(channel, chT);
    k_count<<<dim3((NEDGES + 255) / 256), dim3(256), 0, stream>>>(Hcols, cnt, v2e);
    k_sort3<<<dim3((NVARS + 255) / 256), dim3(256), 0, stream>>>(v2e);

    for (int it = 0; it < NITERS; ++it) {
        k_var_update<<<dim3(BATCH / 128, NVARS / 8), dim3(32, 8), 0, stream>>>(
            chT, msg, v2e, it == 0 ? 1 : 0);
        k_check_update<<<dim3(BATCH / 512, NCHECKS), dim3(128), 0, stream>>>(
            msg, weights, it);
    }

    k_final<<<dim3(BATCH / 32, NVARS / 32), dim3(32, 8), 0, stream>>>(chT, msg, v2e, out);
}